// OxRNAEnergy_36953898615478
// MI455X (gfx1250) — compile-verified
//
#include <hip/hip_runtime.h>
#include <hip/hip_bf16.h>
#include <math.h>

typedef __attribute__((ext_vector_type(2))) float v2f;
typedef __attribute__((ext_vector_type(8))) float v8f;

#define EXC_EPSf 2.0f

// ---------------------------------------------------------------------------
// Wave32 sum via V_WMMA_F32_16X16X4_F32 ones-matrix row-sum trick.
// A (16x4): lane L<16 holds A[L][0]=s (vgpr0), lane L>=16 holds A[L-16][2]=s.
// B = all ones  =>  D[m][n] = s_m + s_{m+16} for every n.
// D layout (16x16 f32): vgpr r, lanes 0-15 -> M=r, lanes 16-31 -> M=r+8.
// Sum of 8 vgprs: lanes<16 = sum_{m=0..7}, lanes>=16 = sum_{m=8..15}.
// One xor-16 shuffle-add yields the full 32-lane sum in every lane.
// Requires EXEC all ones (call with whole wave active).
// ---------------------------------------------------------------------------
__device__ __forceinline__ float wave_sum32(float s) {
#if defined(__gfx1250__) && __has_builtin(__builtin_amdgcn_wmma_f32_16x16x4_f32)
  v2f a; a[0] = s;    a[1] = 0.0f;
  v2f b; b[0] = 1.0f; b[1] = 1.0f;
  v8f c = {0.f, 0.f, 0.f, 0.f, 0.f, 0.f, 0.f, 0.f};
  c = __builtin_amdgcn_wmma_f32_16x16x4_f32(false, a, false, b, (short)0, c,
                                            false, false);
  float t = ((c[0] + c[1]) + (c[2] + c[3])) + ((c[4] + c[5]) + (c[6] + c[7]));
  t += __shfl_xor(t, 16, 32);
  return t;
#else
  for (int m = 16; m >= 1; m >>= 1) s += __shfl_xor(s, m, 32);
  return s;
#endif
}

// ------------------------------- math helpers ------------------------------

__device__ __forceinline__ float f4t(float c, float a, float t0) {
  c = fminf(fmaxf(c, -1.0f + 1e-6f), 1.0f - 1e-6f);
  float t = acosf(c) - t0;
  float v = 1.0f - a * t * t;
  return fminf(fmaxf(v, 0.0f), 1.0f);
}

__device__ __forceinline__ float morse_f1(float r, float eps, float a,
                                          float r0, float rc) {
  if (r >= rc) return 0.0f;
  float w  = 1.0f - expf(-a * (r  - r0));
  float wc = 1.0f - expf(-a * (rc - r0));   // constant-folds per call site
  return eps * ((w * w - 1.0f) - (wc * wc - 1.0f));
}

__device__ __forceinline__ float harm_f2(float r, float k, float r0, float rc,
                                         float rlow) {
  if (r <= rlow || r >= rc) return 0.0f;
  float d = r - r0, dc = rc - r0;
  return 0.5f * k * (d * d - dc * dc);
}

__device__ __forceinline__ float slj(float r, float sigma) {
  float rstar = 0.9f * sigma;
  float rc    = 1.1f * sigma;
  r = fmaxf(r, 0.1f * sigma);
  if (r >= rc) return 0.0f;
  if (r < rstar) {
    float s2 = (sigma / r) * (sigma / r);
    float s6 = s2 * s2 * s2;
    return 4.0f * EXC_EPSf * (s6 * s6 - s6);
  }
  float q  = sigma / rstar;               // constants -> folded
  float q2 = q * q;
  float q6 = q2 * q2 * q2;
  float ljrs = 4.0f * EXC_EPSf * (q6 * q6 - q6);
  float bcon = ljrs / (EXC_EPSf * (rc - rstar) * (rc - rstar));
  float d = rc - r;
  return EXC_EPSf * bcon * d * d;
}

struct Frame {
  float bkx, bky, bkz;   // back site
  float stx, sty, stz;   // stack site
  float bax, bay, baz;   // base site
  float a1x, a1y, a1z;
  float a3x, a3y, a3z;
};

__device__ __forceinline__ Frame load_frame(float px, float py, float pz,
                                            const float4* __restrict__ quat,
                                            int i) {
  float4 q = quat[i];
  float w = q.x, x = q.y, y = q.z, z = q.w;
  float inv = 1.0f / sqrtf(w * w + x * x + y * y + z * z + 1e-12f);
  w *= inv; x *= inv; y *= inv; z *= inv;
  Frame f;
  f.a1x = w * w + x * x - y * y - z * z;
  f.a1y = 2.0f * (x * y + w * z);
  f.a1z = 2.0f * (x * z - w * y);
  f.a3x = 2.0f * (x * z + w * y);
  f.a3y = 2.0f * (y * z - w * x);
  f.a3z = w * w - x * x - y * y + z * z;
  f.bkx = px - 0.4f * f.a1x + 0.2f * f.a3x;
  f.bky = py - 0.4f * f.a1y + 0.2f * f.a3y;
  f.bkz = pz - 0.4f * f.a1z + 0.2f * f.a3z;
  f.stx = px + 0.34f * f.a1x;
  f.sty = py + 0.34f * f.a1y;
  f.stz = pz + 0.34f * f.a1z;
  f.bax = px + 0.4f * f.a1x;
  f.bay = py + 0.4f * f.a1y;
  f.baz = pz + 0.4f * f.a1z;
  return f;
}

__device__ __forceinline__ float mic(float d, float b, float ib) {
  return d - b * rintf(d * ib);
}

// ------------------------------ bonded kernel ------------------------------

__global__ void __launch_bounds__(256)
bonded_energy_kernel(const float* __restrict__ pos,
                     const float4* __restrict__ quat,
                     const float* __restrict__ box,
                     const int2* __restrict__ pairs, int npairs,
                     const float* __restrict__ st_eps,
                     const float* __restrict__ st_shift,
                     float* __restrict__ partials) {
  __shared__ float s_wave[8];
  const int t = threadIdx.x;
  const float bx = box[0], by = box[1], bz = box[2];
  const float ibx = 1.0f / bx, iby = 1.0f / by, ibz = 1.0f / bz;
  float e = 0.0f;
  const int idx = blockIdx.x * 256 + t;
  if (idx < npairs) {
    int2 p = pairs[idx];
    int i = p.x, j = p.y;
    float pix = pos[3 * i], piy = pos[3 * i + 1], piz = pos[3 * i + 2];
    float pjx = pos[3 * j], pjy = pos[3 * j + 1], pjz = pos[3 * j + 2];
    Frame fi = load_frame(pix, piy, piz, quat, i);
    Frame fj = load_frame(pjx, pjy, pjz, quat, j);

    // FENE on back-back distance
    {
      float dx = mic(fj.bkx - fi.bkx, bx, ibx);
      float dy = mic(fj.bky - fi.bky, by, iby);
      float dz = mic(fj.bkz - fi.bkz, bz, ibz);
      float r = sqrtf(dx * dx + dy * dy + dz * dz + 1e-12f);
      float d0 = r - 0.761524f;
      float x2 = fminf(d0 * d0 / 0.04f, 0.999f);
      e += -0.5f * 2.0f * log1pf(-x2);
    }
    // bonded excluded volume: (base_i,base_j,SS) (base_i,back_j,BS) (back_i,base_j,BS)
    {
      float dx = mic(fj.bax - fi.bax, bx, ibx);
      float dy = mic(fj.bay - fi.bay, by, iby);
      float dz = mic(fj.baz - fi.baz, bz, ibz);
      e += slj(sqrtf(dx * dx + dy * dy + dz * dz + 1e-12f), 0.33f);
      dx = mic(fj.bkx - fi.bax, bx, ibx);
      dy = mic(fj.bky - fi.bay, by, iby);
      dz = mic(fj.bkz - fi.baz, bz, ibz);
      e += slj(sqrtf(dx * dx + dy * dy + dz * dz + 1e-12f), 0.515f);
      dx = mic(fj.bax - fi.bkx, bx, ibx);
      dy = mic(fj.bay - fi.bky, by, iby);
      dz = mic(fj.baz - fi.bkz, bz, ibz);
      e += slj(sqrtf(dx * dx + dy * dy + dz * dz + 1e-12f), 0.515f);
    }
    // stacking
    {
      float dx = mic(fj.stx - fi.stx, bx, ibx);
      float dy = mic(fj.sty - fi.sty, by, iby);
      float dz = mic(fj.stz - fi.stz, bz, ibz);
      float r = sqrtf(dx * dx + dy * dy + dz * dz + 1e-12f);
      float rhx = dx / r, rhy = dy / r, rhz = dz / r;
      float f1s = morse_f1(r, st_eps[idx], 6.0f, 0.4f, 0.9f)
                - ((r < 0.9f) ? st_shift[idx] : 0.0f);
      float a33 = fi.a3x * fj.a3x + fi.a3y * fj.a3y + fi.a3z * fj.a3z;
      float ri  = rhx * fi.a3x + rhy * fi.a3y + rhz * fi.a3z;
      float rj  = rhx * fj.a3x + rhy * fj.a3y + rhz * fj.a3z;
      float ang = f4t(a33, 1.3f, 0.0f) * f4t(ri, 0.9f, 0.0f)
                * f4t(-rj, 0.9f, 0.0f);
      e += f1s * ang;
    }
  }
  float wsum = wave_sum32(e);
  if ((t & 31) == 0) s_wave[t >> 5] = wsum;
  __syncthreads();
  if (t == 0) {
    float b = 0.0f;
#pragma unroll
    for (int k = 0; k < 8; ++k) b += s_wave[k];
    partials[blockIdx.x] = b;
  }
}

// ---------------------------- nonbonded kernel -----------------------------

__global__ void __launch_bounds__(256)
nonbonded_energy_kernel(const float* __restrict__ pos,
                        const float4* __restrict__ quat,
                        const float* __restrict__ box,
                        const int2* __restrict__ pairs, int npairs,
                        const int* __restrict__ types,
                        const float* __restrict__ hbeps,
                        const float* __restrict__ crk,
                        float* __restrict__ partials) {
  __shared__ float s_hb[16], s_cr[16], s_wave[8];
  const int t = threadIdx.x;
  if (t < 16) { s_hb[t] = hbeps[t]; s_cr[t] = crk[t]; }
  __syncthreads();
  const float bx = box[0], by = box[1], bz = box[2];
  const float ibx = 1.0f / bx, iby = 1.0f / by, ibz = 1.0f / bz;
  float e = 0.0f;
  const int idx = blockIdx.x * 256 + t;
  if (idx < npairs) {
    int2 p = pairs[idx];
    int i = p.x, j = p.y;
    float pix = pos[3 * i], piy = pos[3 * i + 1], piz = pos[3 * i + 2];
    float pjx = pos[3 * j], pjy = pos[3 * j + 1], pjz = pos[3 * j + 2];
    float dcx = mic(pjx - pix, bx, ibx);
    float dcy = mic(pjy - piy, by, iby);
    float dcz = mic(pjz - piz, bz, ibz);
    // All site-level cutoffs (max 0.77) + max site offset (2*0.447) < 2.0:
    // beyond that every term is exactly zero.
    if (dcx * dcx + dcy * dcy + dcz * dcz < 4.0f) {
      Frame fi = load_frame(pix, piy, piz, quat, i);
      Frame fj = load_frame(pjx, pjy, pjz, quat, j);

      // base_j - base_i (shared by SS-excl, HB, cross-stacking)
      float hbx = mic(fj.bax - fi.bax, bx, ibx);
      float hby = mic(fj.bay - fi.bay, by, iby);
      float hbz = mic(fj.baz - fi.baz, bz, ibz);
      float rhb = sqrtf(hbx * hbx + hby * hby + hbz * hbz + 1e-12f);
      e += slj(rhb, 0.33f);
      // base_i -> back_j
      {
        float dx = mic(fj.bkx - fi.bax, bx, ibx);
        float dy = mic(fj.bky - fi.bay, by, iby);
        float dz = mic(fj.bkz - fi.baz, bz, ibz);
        e += slj(sqrtf(dx * dx + dy * dy + dz * dz + 1e-12f), 0.515f);
      }
      // back_i -> base_j
      {
        float dx = mic(fj.bax - fi.bkx, bx, ibx);
        float dy = mic(fj.bay - fi.bky, by, iby);
        float dz = mic(fj.baz - fi.bkz, bz, ibz);
        e += slj(sqrtf(dx * dx + dy * dy + dz * dz + 1e-12f), 0.515f);
      }
      // back-back
      {
        float dx = mic(fj.bkx - fi.bkx, bx, ibx);
        float dy = mic(fj.bky - fi.bky, by, iby);
        float dz = mic(fj.bkz - fi.bkz, bz, ibz);
        e += slj(sqrtf(dx * dx + dy * dy + dz * dz + 1e-12f), 0.7f);
      }

      float rx = hbx / rhb, ry = hby / rhb, rz = hbz / rhb;
      int ti = types[i], tj = types[j];
      int ssum = ti + tj;
      float comp = (ssum == 3 || ssum == 5) ? 1.0f : 0.0f;
      float eps_hb = s_hb[ti * 4 + tj] * comp;

      float a11 = fi.a1x * fj.a1x + fi.a1y * fj.a1y + fi.a1z * fj.a1z;
      float a33 = fi.a3x * fj.a3x + fi.a3y * fj.a3y + fi.a3z * fj.a3z;
      float r_a1i = rx * fi.a1x + ry * fi.a1y + rz * fi.a1z;
      float r_a1j = rx * fj.a1x + ry * fj.a1y + rz * fj.a1z;
      float r_a3i = rx * fi.a3x + ry * fi.a3y + rz * fi.a3z;
      float r_a3j = rx * fj.a3x + ry * fj.a3y + rz * fj.a3z;

      // hydrogen bonding
      float ang_hb = f4t(-a11, 1.5f, 0.0f) * f4t(r_a1i, 1.5f, 0.0f)
                   * f4t(-r_a1j, 1.5f, 0.0f) * f4t(a33, 0.46f, 0.0f);
      e += morse_f1(rhb, eps_hb, 8.0f, 0.4f, 0.75f) * ang_hb;

      // cross-stacking
      float kcr = s_cr[ti * 4 + tj];
      float ang_cr = f4t(a33, 2.25f, 0.791592f) * f4t(r_a3i, 1.7f, 0.0f)
                   * f4t(-r_a3j, 1.7f, 0.0f);
      e += harm_f2(rhb, kcr, 0.575f, 0.675f, 0.495f) * ang_cr;

      // coaxial stacking (stack sites)
      {
        float dx = mic(fj.stx - fi.stx, bx, ibx);
        float dy = mic(fj.sty - fi.sty, by, iby);
        float dz = mic(fj.stz - fi.stz, bz, ibz);
        float rcxr = sqrtf(dx * dx + dy * dy + dz * dz + 1e-12f);
        float cx = dx / rcxr, cy = dy / rcxr, cz = dz / rcxr;
        float c_a3i = cx * fi.a3x + cy * fi.a3y + cz * fi.a3z;
        float c_a3j = cx * fj.a3x + cy * fj.a3y + cz * fj.a3z;
        float ang_cx = f4t(-a33, 2.0f, 0.0f) * f4t(c_a3i, 1.3f, 0.0f)
                     * f4t(c_a3j, 1.3f, 0.0f);
        e += harm_f2(rcxr, 46.0f, 0.4f, 0.6f, 0.22f) * ang_cx;
      }
    }
  }
  float wsum = wave_sum32(e);
  if ((t & 31) == 0) s_wave[t >> 5] = wsum;
  __syncthreads();
  if (t == 0) {
    float b = 0.0f;
#pragma unroll
    for (int k = 0; k < 8; ++k) b += s_wave[k];
    partials[blockIdx.x] = b;
  }
}

// ------------------------------ final reduce -------------------------------

__global__ void __launch_bounds__(256)
final_reduce_kernel(const float* __restrict__ partials, int n,
                    float* __restrict__ out) {
  __shared__ float s_wave[8];
  const int t = threadIdx.x;
  float e = 0.0f;
  for (int k = t; k < n; k += 256) e += partials[k];
  float wsum = wave_sum32(e);
  if ((t & 31) == 0) s_wave[t >> 5] = wsum;
  __syncthreads();
  if (t == 0) {
    float b = 0.0f;
#pragma unroll
    for (int k = 0; k < 8; ++k) b += s_wave[k];
    out[0] = b;
  }
}

// -------------------------------- launcher ---------------------------------

extern "C" void kernel_launch(void* const* d_in, const int* in_sizes, int n_in,
                              void* d_out, int out_size, void* d_ws,
                              size_t ws_size, hipStream_t stream) {
  const float*  pos    = (const float*)d_in[0];
  const float4* quat   = (const float4*)d_in[1];
  const float*  box    = (const float*)d_in[2];
  const int2*   bp     = (const int2*)d_in[3];
  const int2*   nbp    = (const int2*)d_in[4];
  const int*    types  = (const int*)d_in[5];
  const float*  steps  = (const float*)d_in[6];
  const float*  sshift = (const float*)d_in[7];
  const float*  hbeps  = (const float*)d_in[8];
  const float*  crossk = (const float*)d_in[9];

  const int nb  = in_sizes[3] / 2;
  const int nnb = in_sizes[4] / 2;
  const int PB  = (nb + 255) / 256;
  const int PN  = (nnb + 255) / 256;

  float* partials = (float*)d_ws;

  bonded_energy_kernel<<<PB, 256, 0, stream>>>(pos, quat, box, bp, nb, steps,
                                               sshift, partials);
  nonbonded_energy_kernel<<<PN, 256, 0, stream>>>(pos, quat, box, nbp, nnb,
                                                  types, hbeps, crossk,
                                                  partials + PB);
  final_reduce_kernel<<<1, 256, 0, stream>>>(partials, PB + PN, (float*)d_out);
}